// Net_35364760715853
// MI455X (gfx1250) — compile-verified
//
#include <hip/hip_runtime.h>
#include <stdint.h>
#include <math.h>

// ---------------------------------------------------------------------------
// GCN 2-layer forward for MI455X (gfx1250, wave32, WMMA).
//   x:[N,512] f32, edge_index:[2,E] i64, W1:[512,64], b1:[64], W2:[64,16], b2:[16]
//   out:[N,16] f32 log-softmax
// ---------------------------------------------------------------------------

#define D_FEAT  512
#define HIDDEN  64
#define NCLS    16

// LDS row stride: 68 floats = 272 B. Divisible by 8 (aligned v2f loads) and
// 68 mod 64 = 4 -> lanes 0-15 hit banks {4n}, lanes 16-31 (k+2) hit {4n+2}:
// conflict-free fragment reads.
#define LPAD 68

typedef __attribute__((ext_vector_type(2))) float v2f;
typedef __attribute__((ext_vector_type(8))) float v8f;
typedef int v4i __attribute__((vector_size(16)));            // matches builtin proto
typedef __attribute__((address_space(3))) v4i lds_v4i;

#if defined(__gfx1250__) && __has_builtin(__builtin_amdgcn_global_load_async_to_lds_b128)
#define HAVE_ASYNC_LDS 1
#else
#define HAVE_ASYNC_LDS 0
#endif

// 16-byte global -> LDS copy; async (no VGPR round-trip, ASYNCcnt) when available.
__device__ __forceinline__ void copy16_g2lds(const float* __restrict__ g, float* l) {
#if HAVE_ASYNC_LDS
    __builtin_amdgcn_global_load_async_to_lds_b128(
        (v4i*)(uintptr_t)g,
        (lds_v4i*)(uintptr_t)l,
        0, 0);
#else
    *(float4*)l = *(const float4*)g;
#endif
}

__device__ __forceinline__ void stage_fence() {
#if HAVE_ASYNC_LDS
#if __has_builtin(__builtin_amdgcn_s_wait_asynccnt)
    __builtin_amdgcn_s_wait_asynccnt(0);
#else
    asm volatile("s_wait_asynccnt 0x0" ::: "memory");
#endif
#endif
    __syncthreads();
}

// ---------------- degree / normalization ----------------

__global__ void k_deg_init(float* __restrict__ deg, int n) {
    int i = blockIdx.x * blockDim.x + threadIdx.x;
    if (i < n) deg[i] = 1.0f;               // self-loop contributes 1
}

__global__ void k_deg_count(const long long* __restrict__ dst, float* __restrict__ deg,
                            long long E) {
    long long i = (long long)blockIdx.x * blockDim.x + threadIdx.x;
    long long stride = (long long)gridDim.x * blockDim.x;
    for (; i < E; i += stride)
        atomicAdd(&deg[dst[i]], 1.0f);
}

__global__ void k_dinv(float* __restrict__ deg, int n) {
    int i = blockIdx.x * blockDim.x + threadIdx.x;
    if (i < n) {
        float d = deg[i];
        deg[i] = (d > 0.0f) ? rsqrtf(d) : 0.0f;   // in place: deg -> dinv
    }
}

// ---------------- GEMM1: h1 = x @ W1  (N x 512) @ (512 x 64) ----------------
// 256 threads = 8 waves; each wave computes a 16x64 tile (4 N-subtiles of 16).
// K streamed through LDS in chunks of 64. W tile stored TRANSPOSED so both
// fragment components are a contiguous, aligned 64-bit LDS load.

__global__ __launch_bounds__(256)
void k_gemm1(const float* __restrict__ x, const float* __restrict__ W1,
             float* __restrict__ h1, int nrows) {
    __shared__ float sX[128][LPAD];    // x tile, row-major [m][k]
    __shared__ float sWt[64][LPAD];    // W tile, transposed [n][k]

    const int tid     = threadIdx.x;
    const int lane    = tid & 31;
    const int wv      = tid >> 5;
    const int halfsel = lane >> 4;     // 0: lanes 0-15, 1: lanes 16-31
    const int l16     = lane & 15;
    const int rowBase = blockIdx.x * 128;

    v8f acc[4] = {};

    for (int kc = 0; kc < D_FEAT; kc += 64) {
        // stage 128x64 tile of x (coalesced 16B, async->LDS when available)
        #pragma unroll
        for (int i = 0; i < 8; ++i) {
            int linear = i * 256 + tid;          // float4 slot in 128x16 grid
            int r  = linear >> 4;
            int c4 = linear & 15;
            int gr = rowBase + r;
            if (gr >= nrows) gr = nrows - 1;     // clamp: keep EXEC uniform
            copy16_g2lds(x + (size_t)gr * D_FEAT + kc + c4 * 4, &sX[r][c4 * 4]);
        }
        // stage 64x64 tile of W1, transposed (coalesced global read)
        #pragma unroll
        for (int i = 0; i < 4; ++i) {
            int linear = i * 256 + tid;          // float4 slot in 64x16 grid
            int r  = linear >> 4;                // k within chunk
            int c4 = linear & 15;                // n/4
            float4 v = *(const float4*)(W1 + (size_t)(kc + r) * HIDDEN + c4 * 4);
            sWt[c4 * 4 + 0][r] = v.x; sWt[c4 * 4 + 1][r] = v.y;
            sWt[c4 * 4 + 2][r] = v.z; sWt[c4 * 4 + 3][r] = v.w;
        }
        stage_fence();

        const float* aRow = &sX[wv * 16 + l16][0];
#if __has_builtin(__builtin_amdgcn_wmma_f32_16x16x4_f32)
        #pragma unroll
        for (int k = 0; k < 64; k += 4) {
            // A 16x4 f32 frag (ISA 7.12.2): lanes 0-15 K=k..k+1, lanes 16-31 K=k+2..k+3
            v2f a = *(const v2f*)(aRow + k + 2 * halfsel);
            #pragma unroll
            for (int nt = 0; nt < 4; ++nt) {
                v2f b = *(const v2f*)(&sWt[nt * 16 + l16][k + 2 * halfsel]);
                acc[nt] = __builtin_amdgcn_wmma_f32_16x16x4_f32(
                    false, a, false, b, (short)0, acc[nt], false, false);
            }
        }
#else
        for (int k = 0; k < 64; k += 4) {
            #pragma unroll
            for (int nt = 0; nt < 4; ++nt)
                #pragma unroll
                for (int g = 0; g < 8; ++g) {
                    int m = g + 8 * halfsel;
                    float s = 0.f;
                    #pragma unroll
                    for (int kk = 0; kk < 4; ++kk)
                        s += sX[wv * 16 + m][k + kk] * sWt[nt * 16 + l16][k + kk];
                    acc[nt][g] += s;
                }
        }
#endif
        __syncthreads();
    }

    // store D: VGPR g -> M = g (+8 for lanes 16-31), N = lane&15
    #pragma unroll
    for (int nt = 0; nt < 4; ++nt)
        #pragma unroll
        for (int g = 0; g < 8; ++g) {
            int m  = g + 8 * halfsel;
            int gr = rowBase + wv * 16 + m;
            if (gr < nrows)
                h1[(size_t)gr * HIDDEN + nt * 16 + l16] = acc[nt][g];
        }
}

// ---------------- GEMM2: h2 = agg1 @ W2  (N x 64) @ (64 x 16) ----------------

__global__ __launch_bounds__(256)
void k_gemm2(const float* __restrict__ A, const float* __restrict__ W2,
             float* __restrict__ h2, int nrows) {
    __shared__ float sA[128][LPAD];
    __shared__ float sW2t[16][LPAD];   // transposed [n][k]

    const int tid     = threadIdx.x;
    const int lane    = tid & 31;
    const int wv      = tid >> 5;
    const int halfsel = lane >> 4;
    const int l16     = lane & 15;
    const int rowBase = blockIdx.x * 128;

    v8f acc = {};

    #pragma unroll
    for (int i = 0; i < 8; ++i) {
        int linear = i * 256 + tid;
        int r  = linear >> 4;
        int c4 = linear & 15;
        int gr = rowBase + r;
        if (gr >= nrows) gr = nrows - 1;
        copy16_g2lds(A + (size_t)gr * HIDDEN + c4 * 4, &sA[r][c4 * 4]);
    }
    {
        // 64x16 W2, transposed into LDS (coalesced global read)
        #pragma unroll
        for (int i = 0; i < 4; ++i) {
            int e = i * 256 + tid;     // 0..1023
            int r = e >> 4;            // k 0..63
            int c = e & 15;            // n 0..15
            sW2t[c][r] = W2[(size_t)r * NCLS + c];
        }
    }
    stage_fence();

    const float* aRow = &sA[wv * 16 + l16][0];
#if __has_builtin(__builtin_amdgcn_wmma_f32_16x16x4_f32)
    #pragma unroll
    for (int k = 0; k < 64; k += 4) {
        v2f a = *(const v2f*)(aRow + k + 2 * halfsel);
        v2f b = *(const v2f*)(&sW2t[l16][k + 2 * halfsel]);
        acc = __builtin_amdgcn_wmma_f32_16x16x4_f32(
            false, a, false, b, (short)0, acc, false, false);
    }
#else
    for (int k = 0; k < 64; k += 4)
        #pragma unroll
        for (int g = 0; g < 8; ++g) {
            int m = g + 8 * halfsel;
            float s = 0.f;
            #pragma unroll
            for (int kk = 0; kk < 4; ++kk)
                s += sA[wv * 16 + m][k + kk] * sW2t[l16][k + kk];
            acc[g] += s;
        }
#endif

    #pragma unroll
    for (int g = 0; g < 8; ++g) {
        int m  = g + 8 * halfsel;
        int gr = rowBase + wv * 16 + m;
        if (gr < nrows)
            h2[(size_t)gr * NCLS + l16] = acc[g];
    }
}

// ---------------- aggregation ----------------

// out[i,:] = h[i,:] * dinv[i]^2   (self-loop term; also initializes every element of out)
__global__ void k_selfloop(const float* __restrict__ h, const float* __restrict__ dinv,
                           float* __restrict__ out, int n, int F) {
    long long idx = (long long)blockIdx.x * blockDim.x + threadIdx.x;
    long long total = (long long)n * F;
    if (idx < total) {
        int node = (int)(idx / F);
        float c = dinv[node];
        out[idx] = h[idx] * c * c;
    }
}

// wave-per-edge scatter, F = 64 (lane carries float2)
__global__ void k_agg_edges64(const float* __restrict__ h,
                              const long long* __restrict__ src,
                              const long long* __restrict__ dst,
                              const float* __restrict__ dinv,
                              float* __restrict__ out, long long E) {
    const int lane = threadIdx.x & 31;
    long long wave  = (long long)blockIdx.x * (blockDim.x >> 5) + (threadIdx.x >> 5);
    long long nwave = (long long)gridDim.x * (blockDim.x >> 5);
    for (long long e = wave; e < E; e += nwave) {
        long long s = src[e];
        long long d = dst[e];
        if (e + nwave < E)   // hint next edge's rows toward L2 (global_prefetch_b8)
            __builtin_prefetch(h + (size_t)src[e + nwave] * HIDDEN, 0, 1);
        float c = dinv[s] * dinv[d];
        float2 v = ((const float2*)(h + (size_t)s * HIDDEN))[lane];
        atomicAdd(&out[(size_t)d * HIDDEN + lane * 2 + 0], v.x * c);
        atomicAdd(&out[(size_t)d * HIDDEN + lane * 2 + 1], v.y * c);
    }
}

// thread-per-(edge,class) scatter, F = 16
__global__ void k_agg_edges16(const float* __restrict__ h,
                              const long long* __restrict__ src,
                              const long long* __restrict__ dst,
                              const float* __restrict__ dinv,
                              float* __restrict__ out, long long E) {
    long long idx = (long long)blockIdx.x * blockDim.x + threadIdx.x;
    long long total = E * NCLS;
    long long stride = (long long)gridDim.x * blockDim.x;
    for (; idx < total; idx += stride) {
        long long e = idx >> 4;
        int f = (int)(idx & 15);
        long long s = src[e];
        long long d = dst[e];
        float c = dinv[s] * dinv[d];
        atomicAdd(&out[(size_t)d * NCLS + f], h[(size_t)s * NCLS + f] * c);
    }
}

__global__ void k_bias_relu(float* __restrict__ a, const float* __restrict__ b, int n) {
    long long idx = (long long)blockIdx.x * blockDim.x + threadIdx.x;
    long long total = (long long)n * HIDDEN;
    if (idx < total) {
        float v = a[idx] + b[idx & (HIDDEN - 1)];
        a[idx] = v > 0.0f ? v : 0.0f;
    }
}

// out[r,:] = log_softmax(agg[r,:] + b2)
__global__ void k_logsoftmax(const float* __restrict__ agg, const float* __restrict__ b2,
                             float* __restrict__ out, int n) {
    int r = blockIdx.x * blockDim.x + threadIdx.x;
    if (r >= n) return;
    float v[NCLS];
    float m = -INFINITY;
    #pragma unroll
    for (int i = 0; i < NCLS; ++i) {
        v[i] = agg[(size_t)r * NCLS + i] + b2[i];
        m = fmaxf(m, v[i]);
    }
    float s = 0.0f;
    #pragma unroll
    for (int i = 0; i < NCLS; ++i) s += __expf(v[i] - m);
    float ls = __logf(s) + m;
    #pragma unroll
    for (int i = 0; i < NCLS; ++i) out[(size_t)r * NCLS + i] = v[i] - ls;
}

// ---------------- host side ----------------

extern "C" void kernel_launch(void* const* d_in, const int* in_sizes, int n_in,
                              void* d_out, int out_size, void* d_ws, size_t ws_size,
                              hipStream_t stream) {
    const float*      x   = (const float*)d_in[0];
    const long long*  ei  = (const long long*)d_in[1];
    const float*      W1  = (const float*)d_in[2];
    const float*      b1  = (const float*)d_in[3];
    const float*      W2  = (const float*)d_in[4];
    const float*      b2  = (const float*)d_in[5];
    float*            out = (float*)d_out;

    const int       N = in_sizes[0] / D_FEAT;       // 100000
    const long long E = (long long)in_sizes[1] / 2; // 3200000
    const long long* src = ei;
    const long long* dst = ei + E;

    // workspace carve-up (256B aligned)
    char* ws = (char*)d_ws;
    size_t off = 0;
    auto carve = [&](size_t bytes) {
        void* p = ws + off;
        off += (bytes + 255) & ~(size_t)255;
        return p;
    };
    float* dinv = (float*)carve((size_t)N * 4);             //  0.4 MB
    float* h1   = (float*)carve((size_t)N * HIDDEN * 4);    // 25.6 MB
    float* agg1 = (float*)carve((size_t)N * HIDDEN * 4);    // 25.6 MB
    float* h2   = (float*)carve((size_t)N * NCLS * 4);      //  6.4 MB
    float* agg2 = (float*)carve((size_t)N * NCLS * 4);      //  6.4 MB
    (void)ws_size; (void)n_in; (void)out_size;

    const int T = 256;
    const int gemmBlocks = (N + 127) / 128;

    // 1) degrees -> D^-1/2
    k_deg_init <<<(N + T - 1) / T, T, 0, stream>>>(dinv, N);
    k_deg_count<<<2048, T, 0, stream>>>(dst, dinv, E);
    k_dinv     <<<(N + T - 1) / T, T, 0, stream>>>(dinv, N);

    // 2) layer 1
    k_gemm1      <<<gemmBlocks, T, 0, stream>>>(x, W1, h1, N);
    k_selfloop   <<<(int)(((long long)N * HIDDEN + T - 1) / T), T, 0, stream>>>(h1, dinv, agg1, N, HIDDEN);
    k_agg_edges64<<<4096, T, 0, stream>>>(h1, src, dst, dinv, agg1, E);
    k_bias_relu  <<<(int)(((long long)N * HIDDEN + T - 1) / T), T, 0, stream>>>(agg1, b1, N);

    // 3) layer 2
    k_gemm2      <<<gemmBlocks, T, 0, stream>>>(agg1, W2, h2, N);
    k_selfloop   <<<(int)(((long long)N * NCLS + T - 1) / T), T, 0, stream>>>(h2, dinv, agg2, N, NCLS);
    k_agg_edges16<<<8192, T, 0, stream>>>(h2, src, dst, dinv, agg2, E);

    // 4) bias + log-softmax
    k_logsoftmax<<<(N + T - 1) / T, T, 0, stream>>>(agg2, b2, out, N);
}